// GCNModel_41858751267050
// MI455X (gfx1250) — compile-verified
//
#include <hip/hip_runtime.h>
#include <hip/hip_bf16.h>

// CDNA5 / gfx1250: wave32, WMMA f32 16x16x4 for full-fp32 GEMM.
typedef __attribute__((ext_vector_type(2))) float v2f;
typedef __attribute__((ext_vector_type(8))) float v8f;

#define CH 128          // Cin = Ch = Co = 128
#define EPS 1e-5f

// ---------------------------------------------------------------------------
// deg[n] = 1.0 (self loop)
__global__ void k_deg_init(float* __restrict__ deg, int N) {
    int i = blockIdx.x * blockDim.x + threadIdx.x;
    if (i < N) deg[i] = 1.0f;
}

// deg[dst] += 1 for each valid edge (valid_w is 1.0 when indices in range)
__global__ void k_edge_deg(const int* __restrict__ ei, float* __restrict__ deg,
                           int E, int N) {
    int e = blockIdx.x * blockDim.x + threadIdx.x;
    if (e >= E) return;
    int s = ei[e];
    int d = ei[E + e];
    if ((unsigned)s < (unsigned)N && (unsigned)d < (unsigned)N) {
        atomicAdd(&deg[d], 1.0f);
    }
}

// dinv[n] = 1/sqrt(deg[n])   (deg >= 1 always)
__global__ void k_dinv(float* __restrict__ deg, int N) {
    int i = blockIdx.x * blockDim.x + threadIdx.x;
    if (i < N) deg[i] = rsqrtf(deg[i]);
}

// ---------------------------------------------------------------------------
// H[M x 128] = X[M x 128] @ W[128 x 128] via v_wmma_f32_16x16x4_f32.
//
// Block = 64 rows x 32 cols, 256 threads = 8 waves arranged 4(m) x 2(n);
// each wave owns one 16x16 output tile, K=128 marched in steps of 4.
//
// LDS staging (≈50.7 KB static):
//   sA[64][132] : x rows, coalesced float4 loads, pad 132 -> conflict-free
//                 ds_load_b64 in the inner loop (bank = 4*row + k).
//   sB[32][132] : W columns transposed (sB[c][k] = W[k][n0+c]) so the B
//                 fragment (two consecutive K values) is one ds_load_b64.
//
// A layout (32-bit A 16x4): lane l holds row M=(l&15); K pairs split across
// lane halves (lanes 0-15 -> k0+0/1, lanes 16-31 -> k0+2/3).
// C/D layout: VGPR r: lanes 0-15 -> M=r, lanes 16-31 -> M=8+r; N = (lane&15).
// Rows >= M are clamped at load; their garbage accumulator rows are never
// stored, so no in-loop masking is needed.
__global__ __launch_bounds__(256) void k_gemm_wmma(const float* __restrict__ X,
                                                   const float* __restrict__ W,
                                                   float* __restrict__ H, int M) {
    __shared__ float sA[64][132];
    __shared__ float sB[32][132];
    const int m0  = blockIdx.x * 64;
    const int n0  = blockIdx.y * 32;
    const int tid = threadIdx.x;

    // Stage A: 64 rows x 128 K = 2048 float4 / 256 threads = 8 each.
    #pragma unroll
    for (int i = 0; i < 8; ++i) {
        const int idx   = tid + i * 256;      // 0..2047
        const int row   = idx >> 5;           // 0..63
        const int chunk = idx & 31;           // float4 index within row
        const int gr    = m0 + row;
        const float4 v =
            *(const float4*)(X + (size_t)(gr < M ? gr : 0) * CH + chunk * 4);
        *(float4*)&sA[row][chunk * 4] = v;
    }
    // Stage B transposed: sB[c][k] = W[k][n0+c]; lanes sweep c (coalesced).
    {
        const int c     = tid & 31;
        const int kbase = (tid >> 5) * 16;
        #pragma unroll
        for (int i = 0; i < 16; ++i) {
            const int k = kbase + i;
            sB[c][k] = W[(size_t)k * CH + n0 + c];
        }
    }
    __syncthreads();

    const int wave  = tid >> 5;
    const int lane  = tid & 31;
    const int mt    = wave >> 1;              // 0..3
    const int nt    = wave & 1;               // 0..1
    const int mloc  = lane & 15;
    const int khalf = (lane >> 4) * 2;        // 0 or 2

    const float* arow = &sA[mt * 16 + mloc][0];
    const float* brow = &sB[nt * 16 + mloc][0];

    v8f c = {};
    #pragma unroll 8
    for (int k0 = 0; k0 < CH; k0 += 4) {
        const v2f a = *(const v2f*)(arow + k0 + khalf);   // ds_load_b64
        const v2f b = *(const v2f*)(brow + k0 + khalf);   // ds_load_b64
        c = __builtin_amdgcn_wmma_f32_16x16x4_f32(
                /*neg_a=*/false, a, /*neg_b=*/false, b,
                /*c_mod=*/(short)0, c, /*reuse_a=*/false, /*reuse_b=*/false);
    }

    const int col   = n0 + nt * 16 + mloc;
    const int half  = lane >> 4;              // 0 -> M=r, 1 -> M=8+r
    const int mbase = m0 + mt * 16 + half * 8;
    if (m0 + 64 <= M) {                       // interior fast path (no branches)
        #pragma unroll
        for (int r = 0; r < 8; ++r)
            H[(size_t)(mbase + r) * CH + col] = c[r];
    } else {
        #pragma unroll
        for (int r = 0; r < 8; ++r)
            if (mbase + r < M) H[(size_t)(mbase + r) * CH + col] = c[r];
    }
}

// ---------------------------------------------------------------------------
// Acc[n,:] = H[n,:] * dinv[n]^2 + bias   (self-loop term; also initializes Acc)
__global__ void k_selfloop_init(const float* __restrict__ H,
                                const float* __restrict__ dinv,
                                const float* __restrict__ bias,
                                float* __restrict__ Acc, int M) {
    int idx = blockIdx.x * blockDim.x + threadIdx.x;   // one float4 per thread
    if (idx >= M * (CH / 4)) return;
    const int row = idx >> 5;        // CH/4 == 32 chunks per row
    const int l   = idx & 31;
    const float di = dinv[row];
    const float w  = di * di;
    float4 h = ((const float4*)(H + (size_t)row * CH))[l];
    float4 b = ((const float4*)bias)[l];
    float4 o;
    o.x = h.x * w + b.x;  o.y = h.y * w + b.y;
    o.z = h.z * w + b.z;  o.w = h.w * w + b.w;
    ((float4*)(Acc + (size_t)row * CH))[l] = o;
}

// One wave per edge; lane owns channels [4*lane, 4*lane+4).
// Acc[dst,:] += H[src,:] * dinv[src]*dinv[dst]
__global__ void k_edge_scatter(const float* __restrict__ H,
                               const int* __restrict__ ei,
                               const float* __restrict__ dinv,
                               float* __restrict__ Acc, int E, int N) {
    int e    = (blockIdx.x * blockDim.x + threadIdx.x) >> 5;
    int lane = threadIdx.x & 31;
    if (e >= E) return;
    int s = ei[e];
    int d = ei[E + e];
    if ((unsigned)s >= (unsigned)N || (unsigned)d >= (unsigned)N) return;
    const float norm = dinv[s] * dinv[d];
    float4 v = ((const float4*)(H + (size_t)s * CH))[lane];
    float* out = Acc + (size_t)d * CH + lane * 4;
    atomicAdd(out + 0, v.x * norm);
    atomicAdd(out + 1, v.y * norm);
    atomicAdd(out + 2, v.z * norm);
    atomicAdd(out + 3, v.w * norm);
}

// ---------------------------------------------------------------------------
// Per-row instance norm (over 128 channels) + ReLU. One wave32 per row.
__global__ void k_rownorm_relu(const float* __restrict__ In,
                               float* __restrict__ Out, int M) {
    int row  = (blockIdx.x * blockDim.x + threadIdx.x) >> 5;
    int lane = threadIdx.x & 31;
    if (row >= M) return;
    float4 v = ((const float4*)(In + (size_t)row * CH))[lane];

    float s = v.x + v.y + v.z + v.w;
    #pragma unroll
    for (int o = 16; o > 0; o >>= 1) s += __shfl_xor(s, o, 32);
    const float mu = s * (1.0f / CH);

    const float d0 = v.x - mu, d1 = v.y - mu, d2 = v.z - mu, d3 = v.w - mu;
    float q = d0 * d0 + d1 * d1 + d2 * d2 + d3 * d3;
    #pragma unroll
    for (int o = 16; o > 0; o >>= 1) q += __shfl_xor(q, o, 32);
    const float inv = rsqrtf(q * (1.0f / CH) + EPS);

    float4 r;
    r.x = fmaxf(d0 * inv, 0.0f);
    r.y = fmaxf(d1 * inv, 0.0f);
    r.z = fmaxf(d2 * inv, 0.0f);
    r.w = fmaxf(d3 * inv, 0.0f);
    ((float4*)(Out + (size_t)row * CH))[lane] = r;
}

// ---------------------------------------------------------------------------
extern "C" void kernel_launch(void* const* d_in, const int* in_sizes, int n_in,
                              void* d_out, int out_size, void* d_ws, size_t ws_size,
                              hipStream_t stream) {
    const float* x   = (const float*)d_in[0];
    const int*   ei  = (const int*)  d_in[1];   // [2, E] (src row, dst row)
    const float* W1  = (const float*)d_in[2];
    const float* b1  = (const float*)d_in[3];
    const float* W2  = (const float*)d_in[4];
    const float* b2  = (const float*)d_in[5];
    float*       out = (float*)d_out;

    const int N = in_sizes[0] / CH;   // 50000 nodes
    const int E = in_sizes[1] / 2;    // 800000 edges

    // Workspace layout (floats): [dinv: Npad][hbuf: N*128][acc: N*128]  ~51.4MB
    float* ws   = (float*)d_ws;
    const size_t nPad = ((size_t)N + 63) & ~(size_t)63;
    float* dinv = ws;                 // used as deg, then overwritten with 1/sqrt
    float* hbuf = ws + nPad;
    float* acc  = hbuf + (size_t)N * CH;

    const int T = 256;
    const int gN     = (N + T - 1) / T;
    const int gE     = (E + T - 1) / T;
    const int gRow4  = (N * (CH / 4) + T - 1) / T;      // one float4/thread
    const int gEdgeW = ((E * 32) + T - 1) / T;          // one wave/edge
    const int gRowW  = ((N * 32) + T - 1) / T;          // one wave/row
    const dim3 gGemm((N + 63) / 64, CH / 32);           // 64-row x 32-col tiles

    // Degree + symmetric normalization
    k_deg_init<<<gN, T, 0, stream>>>(dinv, N);
    k_edge_deg<<<gE, T, 0, stream>>>(ei, dinv, E, N);
    k_dinv   <<<gN, T, 0, stream>>>(dinv, N);

    // ---- Layer 1: h = x@W1 ; acc = scatter(h) + h*dinv^2 + b1 ; norm+relu
    k_gemm_wmma    <<<gGemm, T, 0, stream>>>(x, W1, hbuf, N);
    k_selfloop_init<<<gRow4, T, 0, stream>>>(hbuf, dinv, b1, acc, N);
    k_edge_scatter <<<gEdgeW, T, 0, stream>>>(hbuf, ei, dinv, acc, E, N);
    k_rownorm_relu <<<gRowW, T, 0, stream>>>(acc, hbuf, N);   // hbuf <- normed

    // ---- Layer 2: h2 = hbuf@W2 ; out = scatter(h2) + h2*dinv^2 + b2 ; norm+relu
    k_gemm_wmma    <<<gGemm, T, 0, stream>>>(hbuf, W2, acc, N);
    k_selfloop_init<<<gRow4, T, 0, stream>>>(acc, dinv, b2, out, N);
    k_edge_scatter <<<gEdgeW, T, 0, stream>>>(acc, ei, dinv, out, E, N);
    k_rownorm_relu <<<gRowW, T, 0, stream>>>(out, out, N);    // in-place
}